// Smearing_4647154614796
// MI455X (gfx1250) — compile-verified
//
#include <hip/hip_runtime.h>
#include <math.h>

// Problem constants (BS, D, L, N) = (4, 4, 8, 3), SMEAR_STEPS = 1
#define BSN    4
#define DD     4
#define LL     8
#define NSITE  4096                       // 8^4
#define CFG_DOUBLES (BSN*DD*NSITE*9*2)    // complex128 cfgs as interleaved doubles

typedef float v2f __attribute__((ext_vector_type(2)));
typedef float v8f __attribute__((ext_vector_type(8)));

// ---------------- complex double helpers ----------------
struct cd  { double x, y; };
struct f2c { float  x, y; };

__device__ __forceinline__ cd mkc(double a, double b){ cd r; r.x=a; r.y=b; return r; }
__device__ __forceinline__ cd cadd(cd a, cd b){ return mkc(a.x+b.x, a.y+b.y); }
__device__ __forceinline__ cd csub(cd a, cd b){ return mkc(a.x-b.x, a.y-b.y); }
__device__ __forceinline__ cd cmul(cd a, cd b){ return mkc(fma(a.x,b.x,-(a.y*b.y)), fma(a.x,b.y,a.y*b.x)); }
__device__ __forceinline__ cd crs (double s, cd a){ return mkc(s*a.x, s*a.y); }
__device__ __forceinline__ cd cconj(cd a){ return mkc(a.x, -a.y); }
__device__ __forceinline__ cd cneg (cd a){ return mkc(-a.x, -a.y); }

__device__ __forceinline__ f2c caddf(f2c a, f2c b){ f2c r; r.x=a.x+b.x; r.y=a.y+b.y; return r; }
__device__ __forceinline__ f2c csubf(f2c a, f2c b){ f2c r; r.x=a.x-b.x; r.y=a.y-b.y; return r; }
__device__ __forceinline__ f2c cmulf(f2c a, f2c b){ f2c r; r.x=a.x*b.x-a.y*b.y; r.y=a.x*b.y+a.y*b.x; return r; }

// ---------------- 3x3 complex matrices (row-major flat [9]) ----------------
__device__ __forceinline__ void mload(const double* __restrict__ p, cd* A){
  #pragma unroll
  for (int t=0;t<9;++t) A[t] = mkc(p[2*t], p[2*t+1]);
}
__device__ __forceinline__ void mstore(double* __restrict__ p, const cd* A){
  #pragma unroll
  for (int t=0;t<9;++t){ p[2*t]=A[t].x; p[2*t+1]=A[t].y; }
}
__device__ __forceinline__ void mmul(const cd* A, const cd* B, cd* D){
  #pragma unroll
  for (int i=0;i<3;++i)
    #pragma unroll
    for (int j=0;j<3;++j)
      D[i*3+j] = cadd(cadd(cmul(A[i*3+0],B[0+j]), cmul(A[i*3+1],B[3+j])), cmul(A[i*3+2],B[6+j]));
}
__device__ __forceinline__ void mmul_dagB(const cd* A, const cd* B, cd* D){ // A * B^dag
  #pragma unroll
  for (int i=0;i<3;++i)
    #pragma unroll
    for (int j=0;j<3;++j)
      D[i*3+j] = cadd(cadd(cmul(A[i*3+0],cconj(B[j*3+0])),
                           cmul(A[i*3+1],cconj(B[j*3+1]))),
                           cmul(A[i*3+2],cconj(B[j*3+2])));
}
__device__ __forceinline__ void mdagA_mul(const cd* A, const cd* B, cd* D){ // A^dag * B
  #pragma unroll
  for (int i=0;i<3;++i)
    #pragma unroll
    for (int j=0;j<3;++j)
      D[i*3+j] = cadd(cadd(cmul(cconj(A[0*3+i]),B[0+j]),
                           cmul(cconj(A[1*3+i]),B[3+j])),
                           cmul(cconj(A[2*3+i]),B[6+j]));
}

__device__ __forceinline__ int lin4(const int* x){ return ((x[0]*LL + x[1])*LL + x[2])*LL + x[3]; }
__device__ __forceinline__ double* uptr(double* cfgs, int b, int dir, int site){
  return cfgs + (((size_t)b*DD + dir)*NSITE + site)*18;
}

__global__ void zero_dlogj(double* p){ if ((int)threadIdx.x < BSN) p[threadIdx.x] = 0.0; }

// One checkerboard sub-step: update U_mu on parity-eo sites, accumulate log|det J|.
// blockDim = 32 (one wave32); grid covers BS * NSITE/2 sites exactly, so EXEC is
// all-ones at every WMMA call site.
__global__ __launch_bounds__(32) void stout_step(double* __restrict__ cfgs,
                                                 double* __restrict__ dlogJ,
                                                 const double* __restrict__ rho_root,
                                                 int mu, int eo)
{
  // Padded staging so every fragment access is UNCONDITIONAL:
  //  shA[site][slot][col]: slots 0-2 = a_t real, 3-5 = a_t imag, 6-7 = zero; cols 9-15 zero.
  //  shB[site][slot][col]: slots 0-2 = b_t real, 3-5 = b_t imag, 6-7 = zero; cols 9-15 zero.
  //  shJ[site][0..80] = 9x9 Jacobian; entry 81 = dump slot for invalid WMMA-tile lanes.
  __shared__ float shA[32][8][16];     // 16 KB
  __shared__ float shB[32][8][16];     // 16 KB
  __shared__ f2c   shJ[32][82];        // 20.5 KB

  const int lane = threadIdx.x;
  const int tid  = blockIdx.x*32 + lane;
  const int b    = tid >> 11;          // 2048 masked sites per batch
  const int q    = tid & 2047;
  const int x2 = (q>>2)&7, x1 = (q>>5)&7, x0 = (q>>8)&7;
  const int x3 = 2*(q&3) + ((x0+x1+x2+eo)&1);   // enforce (x0+x1+x2+x3)%2 == eo
  int xc[4] = {x0,x1,x2,x3};
  const int site = lin4(xc);

  // zero-fill this thread's staging slabs (vectorized DS stores)
  {
    float4 z4; z4.x=0.f; z4.y=0.f; z4.z=0.f; z4.w=0.f;
    float4* pa = (float4*)&shA[lane][0][0];
    float4* pb = (float4*)&shB[lane][0][0];
    #pragma unroll
    for (int t=0;t<32;++t){ pa[t] = z4; pb[t] = z4; }
  }

  // ---------------- staples C (f64) ----------------
  cd U[9]; mload(uptr(cfgs,b,mu,site), U);
  cd Cst[9];
  #pragma unroll
  for (int t=0;t<9;++t) Cst[t]=mkc(0.0,0.0);

  for (int nu=0; nu<4; ++nu){
    if (nu==mu) continue;
    double rr = rho_root[(eo+2*mu)*4 + nu]; double rho = rr*rr;
    int xp[4]  = {xc[0],xc[1],xc[2],xc[3]}; xp[nu]  = (xp[nu] +1)&7;  // x+nu
    int xm[4]  = {xc[0],xc[1],xc[2],xc[3]}; xm[nu]  = (xm[nu] +7)&7;  // x-nu
    int xpm[4] = {xc[0],xc[1],xc[2],xc[3]}; xpm[mu] = (xpm[mu]+1)&7;  // x+mu
    int xmp[4] = {xm[0],xm[1],xm[2],xm[3]}; xmp[mu] = (xmp[mu]+1)&7;  // x-nu+mu
    cd A[9], Bm[9], Cm[9], T[9], S[9];
    // up = U_nu(x) U_mu(x+nu) U_nu(x+mu)^dag — reads are parity !eo or dir!=mu: in-place safe
    mload(uptr(cfgs,b,nu,site),      A);
    mload(uptr(cfgs,b,mu,lin4(xp)),  Bm);
    mload(uptr(cfgs,b,nu,lin4(xpm)), Cm);
    mmul(A,Bm,T); mmul_dagB(T,Cm,S);
    #pragma unroll
    for (int t=0;t<9;++t) Cst[t]=cadd(Cst[t], crs(rho, S[t]));
    // dn = U_nu(x-nu)^dag U_mu(x-nu) U_nu(x-nu+mu)
    mload(uptr(cfgs,b,nu,lin4(xm)),  A);
    mload(uptr(cfgs,b,mu,lin4(xm)),  Bm);
    mload(uptr(cfgs,b,nu,lin4(xmp)), Cm);
    mdagA_mul(A,Bm,T); mmul(T,Cm,S);
    #pragma unroll
    for (int t=0;t<9;++t) Cst[t]=cadd(Cst[t], crs(rho, S[t]));
  }

  // ---------------- Q = 0.5i(omega^dag - omega) - traceless part ----------------
  cd om[9]; mmul_dagB(Cst, U, om);
  cd dom[9];
  #pragma unroll
  for (int i=0;i<3;++i)
    #pragma unroll
    for (int j=0;j<3;++j) dom[i*3+j] = csub(cconj(om[j*3+i]), om[i*3+j]);
  cd trd = cadd(cadd(dom[0],dom[4]),dom[8]);
  cd Q[9];
  #pragma unroll
  for (int i=0;i<3;++i)
    #pragma unroll
    for (int j=0;j<3;++j){
      cd d = dom[i*3+j];
      cd v = mkc(-0.5*d.y, 0.5*d.x);               // 0.5i * d
      if (i==j){ v.x += (0.5/3.0)*trd.y; v.y -= (0.5/3.0)*trd.x; }  // -(0.5i/3) tr
      Q[i*3+j]=v;
    }
  cd Q2[9]; mmul(Q,Q,Q2);

  // ---------------- Cayley-Hamilton coefficients (f64 scalars) ----------------
  cd c0c = csub( cadd( cmul(Q[0], csub(cmul(Q[4],Q[8]), cmul(Q[5],Q[7]))),
                       cmul(Q[2], csub(cmul(Q[3],Q[7]), cmul(Q[4],Q[6]))) ),
                 cmul(Q[1], csub(cmul(Q[3],Q[8]), cmul(Q[5],Q[6]))) );
  double trq2 = Q2[0].x + Q2[4].x + Q2[8].x;       // real for Hermitian Q
  double c1   = 0.5*trq2;
  bool   pos  = (c0c.x > 0.0);
  double c0a  = sqrt(c0c.x*c0c.x + c0c.y*c0c.y);
  double c1_3 = c1/3.0; if (c1_3 < 0.0) c1_3 = 0.0;
  double c0mx = 2.0*c1_3*sqrt(c1_3);
  double rat  = (c0mx > 1e-300) ? (c0a/c0mx) : 1.0;
  if (rat > 1.0) rat = 1.0;
  double th = acos(rat);
  double u  = sqrt(c1_3)*cos(th/3.0);
  double w  = sqrt(c1>0.0?c1:0.0)*sin(th/3.0);
  double u2=u*u, w2=w*w, cw=cos(w);
  cd eu  = mkc(cos(u), sin(u));
  cd eu2 = cmul(eu,eu);
  cd eum = cconj(eu);
  double xz, xo;
  if (fabs(w) > 0.005){ xz = sin(w)/w; xo = cos(w)/w2 - sin(w)/(w2*w); }
  else { xz = 1.0 - (1.0/6.0)*w2*(1.0 - (1.0/20.0)*w2*(1.0 - (1.0/42.0)*w2));
         xo = -1.0/3.0 + w2*((1.0/30.0) + w2*(-(1.0/840.0) + (1.0/45360.0)*w2)); }

  cd h0 = cadd(crs(u2-w2, eu2), cmul(eum, mkc(8.0*u2*cw, 2.0*u*(3.0*u2+w2)*xz)));
  cd h1 = csub(crs(2.0*u, eu2), cmul(eum, mkc(2.0*u*cw, -(3.0*u2-w2)*xz)));
  cd h2 = csub(eu2,             cmul(eum, mkc(cw, 3.0*u*xz)));
  cd r10= cadd(crs(2.0, cmul(mkc(u, u2-w2), eu2)),
               crs(2.0, cmul(eum, mkc(8.0*u*cw + u*(3.0*u2+w2)*xz, -4.0*u2*cw + (9.0*u2+w2)*xz))));
  cd r11= cadd(crs(2.0, cmul(mkc(1.0, 2.0*u), eu2)),
               cmul(eum, mkc(-2.0*cw - (w2-3.0*u2)*xz, 2.0*u*cw + 6.0*u*xz)));
  cd t12= cmul(eum, mkc(cw - 3.0*xz, 3.0*u*xz));
  cd r12= cadd(mkc(-2.0*eu2.y, 2.0*eu2.x), mkc(-t12.y, t12.x));
  cd t20= cmul(eum, mkc(cw + xz + 3.0*u2*xo, 4.0*u*xz));
  cd r20= cadd(crs(-2.0, eu2), crs(2.0*u, mkc(-t20.y, t20.x)));
  cd t21= cmul(eum, mkc(cw + xz - 3.0*u2*xo, 2.0*u*xz));
  cd r21= mkc(t21.y, -t21.x);                       // -i * t21
  cd r22= cmul(eum, mkc(xz, -3.0*u*xo));

  double den = 9.0*u2 - w2;
  double idn = 1.0/den, id2 = 1.0/(2.0*den*den);
  double v3 = 3.0*u2-w2, v15 = 15.0*u2+w2;
  cd f0 = crs(idn,h0), f1 = crs(idn,h1), f2 = crs(idn,h2);
  cd b10= crs(id2, csub(cadd(crs(2.0*u,r10), crs(v3,r20)), crs(2.0*v15,f0)));
  cd b11= crs(id2, csub(cadd(crs(2.0*u,r11), crs(v3,r21)), crs(2.0*v15,f1)));
  cd b12= crs(id2, csub(cadd(crs(2.0*u,r12), crs(v3,r22)), crs(2.0*v15,f2)));
  cd b20= crs(id2, csub(csub(r10, crs(3.0*u,r20)), crs(24.0*u,f0)));
  cd b21= crs(id2, csub(csub(r11, crs(3.0*u,r21)), crs(24.0*u,f1)));
  cd b22= crs(id2, csub(csub(r12, crs(3.0*u,r22)), crs(24.0*u,f2)));
  if (!pos){
    f0 = cconj(f0);  f1 = cneg(cconj(f1));  f2 = cconj(f2);
    b10= cconj(b10); b11= cneg(cconj(b11)); b12= cconj(b12);
    b20= cneg(cconj(b20)); b21= cconj(b21); b22= cneg(cconj(b22));
  }

  // ---------------- derived matrices ----------------
  cd QU[9], Q2U[9], Un[9], B1U[9], B2U[9], SU[9];
  cd Cd[9], CdQ[9], CdQ2[9], CdU[9], CdQU[9];
  mmul(Q,U,QU); mmul(Q2,U,Q2U);
  #pragma unroll
  for (int t=0;t<9;++t){
    Un[t]  = cadd(cadd(cmul(f0 ,U[t]), cmul(f1 ,QU[t])), cmul(f2 ,Q2U[t]));
    B1U[t] = cadd(cadd(cmul(b10,U[t]), cmul(b11,QU[t])), cmul(b12,Q2U[t]));
    B2U[t] = cadd(cadd(cmul(b20,U[t]), cmul(b21,QU[t])), cmul(b22,Q2U[t]));
    SU[t]  = cadd(cadd(crs(2.0*c1,B2U[t]), cmul(f1,U[t])), crs(2.0, cmul(f2,QU[t])));
  }
  #pragma unroll
  for (int i=0;i<3;++i)
    #pragma unroll
    for (int j=0;j<3;++j) Cd[i*3+j] = cconj(Cst[j*3+i]);
  mmul(Cd,Q,CdQ); mmul(Cd,Q2,CdQ2); mmul(Cd,U,CdU); mmul(CdQ,U,CdQU);

  // updated link (full f64) — in-place safe (no other thread reads this entry)
  mstore(uptr(cfgs,b,mu,site), Un);

  // f32 copies needed after the WMMA phase (structured Jacobian terms)
  f2c m1f[9], gf[9], cduf[9], eqf[9];
  #pragma unroll
  for (int t=0;t<9;++t){
    cd tt = cadd(cmul(f1,CdU[t]), cmul(f2,CdQU[t]));               // f1*CdU + f2*CdQU
    m1f[t].x = (float)(-0.5*tt.y); m1f[t].y = (float)(0.5*tt.x);   // * 0.5i
    cd g  = cmul(f2, Q[t]);
    gf[t].x  = (float)(-0.5*g.y);  gf[t].y  = (float)(0.5*g.x);    // 0.5i*f2*Q
    cduf[t].x = (float)CdU[t].x;   cduf[t].y = (float)CdU[t].y;
    cd e = cadd(cmul(f1,Q[t]), cmul(f2,Q2[t]));
    if (t==0 || t==4 || t==8) e = cadd(e, f0);
    eqf[t].x = (float)e.x; eqf[t].y = (float)e.y;                  // expQ entries
  }

  // stage outer-product vectors (complex coeffs folded into a):
  //   J9x9 += sum_t a_t (x) b_t,  t: 0.5i*B1U(x)CdQ^T, 0.5i*B2U(x)CdQ2^T, -(i/6)*SU(x)Cd^T
  #pragma unroll
  for (int i=0;i<3;++i)
    #pragma unroll
    for (int l=0;l<3;++l){
      int r = i*3+l;
      shA[lane][0][r] = (float)(-0.5*B1U[r].y);      shA[lane][3][r] = (float)( 0.5*B1U[r].x);
      shA[lane][1][r] = (float)(-0.5*B2U[r].y);      shA[lane][4][r] = (float)( 0.5*B2U[r].x);
      shA[lane][2][r] = (float)((1.0/6.0)*SU[r].y);  shA[lane][5][r] = (float)(-(1.0/6.0)*SU[r].x);
    }
  #pragma unroll
  for (int j=0;j<3;++j)
    #pragma unroll
    for (int k=0;k<3;++k){
      int c = j*3+k;
      shB[lane][0][c] = (float)CdQ [k*3+j].x;  shB[lane][3][c] = (float)CdQ [k*3+j].y;
      shB[lane][1][c] = (float)CdQ2[k*3+j].x;  shB[lane][4][c] = (float)CdQ2[k*3+j].y;
      shB[lane][2][c] = (float)Cd  [k*3+j].x;  shB[lane][5][c] = (float)Cd  [k*3+j].y;
    }
  __syncthreads();

  // ---------------- WMMA phase: 3 complex outer products == K=6 f32 matmul ----------------
  // A frag (16x4): lane m=lane&15 is row M; VGPR v holds K = v + 2*(lane>>4).
  // Real pass K pairs: (ar,br) x3 then (ai,-bi) x3 ; imag pass: (ar,bi) x3 then (ai,br) x3.
  // Slot/sign selection is per-lane index arithmetic on zero-padded LDS; the D-tile
  // scatter routes invalid lanes to dump slot 81 — the loop body has no exec divergence.
  {
    const int m = lane & 15, half = lane >> 4;
    const int k1 = 2*half;        // first-fragment K slots: k1, k1+1   (0..3)
    const int k2 = 4 + 2*half;    // second-fragment K slots: k2, k2+1  (4..7; 6,7 zero)
    const int k1b = k1+1, k2b = k2+1;
    // imag-pass slot remap: kI = (k<6) ? (k+3)%6 : k  (slots 6,7 stay on zero padding)
    const int i1  = (k1 +3)%6;
    const int i1b = (k1b+3)%6;
    const int i2  = (k2  < 6) ? (k2 +3)%6 : k2;
    const int i2b = (k2b < 6) ? (k2b+3)%6 : k2b;
    const float sR1b = (k1b >= 3) ? -1.0f : 1.0f;   // sign for real-pass slot k1+1 (k=3 when half=1)

    #pragma unroll 1
    for (int s = 0; s < 32; ++s){
      v2f a1  = { shA[s][k1][m],          shA[s][k1b][m] };
      v2f a2  = { shA[s][k2][m],          shA[s][k2b][m] };
      v2f bR1 = { shB[s][k1][m],          sR1b*shB[s][k1b][m] };
      v2f bR2 = { -shB[s][k2][m],         -shB[s][k2b][m] };     // k>=4: always -bi (or -0)
      v2f bI1 = { shB[s][i1][m],          shB[s][i1b][m] };
      v2f bI2 = { shB[s][i2][m],          shB[s][i2b][m] };
      v8f z8 = {0.f,0.f,0.f,0.f,0.f,0.f,0.f,0.f};
      v8f dr = __builtin_amdgcn_wmma_f32_16x16x4_f32(false, a1, false, bR1, (short)0, z8, false, false);
      dr     = __builtin_amdgcn_wmma_f32_16x16x4_f32(false, a2, false, bR2, (short)0, dr, false, false);
      v8f di = __builtin_amdgcn_wmma_f32_16x16x4_f32(false, a1, false, bI1, (short)0, z8, false, false);
      di     = __builtin_amdgcn_wmma_f32_16x16x4_f32(false, a2, false, bI2, (short)0, di, false, false);
      // D (16x16): VGPR v, lanes 0-15 -> row v, lanes 16-31 -> row v+8; col = lane&15.
      // Branch-free scatter: invalid (row,col) lanes collide harmlessly on dump slot 81.
      #pragma unroll
      for (int v = 0; v < 8; ++v){
        const int row = v + 8*half;
        const int idx = (row < 9 && m < 9) ? (row*9 + m) : 81;
        shJ[s][idx].x = dr[v]; shJ[s][idx].y = di[v];
      }
    }
  }
  __syncthreads();

  // ---------------- structured terms + expQ*delta, then LU log|det| ----------------
  for (int i=0;i<3;++i)
    for (int l=0;l<3;++l)
      #pragma unroll
      for (int j=0;j<3;++j)
        #pragma unroll
        for (int k=0;k<3;++k){
          int r = i*3+l, c = j*3+k;
          f2c v = shJ[lane][r*9+c];
          if (i==j) v = caddf(v, m1f[k*3+l]);
          v = caddf(v, cmulf(gf[i*3+j], cduf[k*3+l]));
          if (k==l) v = caddf(v, eqf[i*3+j]);
          shJ[lane][r*9+c] = v;
        }

  f2c* M = &shJ[lane][0];
  float logdet = 0.0f;
  for (int cc = 0; cc < 9; ++cc){
    int p = cc; float best = M[cc*9+cc].x*M[cc*9+cc].x + M[cc*9+cc].y*M[cc*9+cc].y;
    for (int r = cc+1; r < 9; ++r){
      float nr = M[r*9+cc].x*M[r*9+cc].x + M[r*9+cc].y*M[r*9+cc].y;
      if (nr > best){ best = nr; p = r; }
    }
    if (p != cc)
      for (int t = 0; t < 9; ++t){ f2c tmp = M[cc*9+t]; M[cc*9+t] = M[p*9+t]; M[p*9+t] = tmp; }
    f2c piv = M[cc*9+cc];
    float nr = piv.x*piv.x + piv.y*piv.y;
    logdet += 0.5f*logf(nr);
    float inv = 1.0f/nr;
    f2c pinv; pinv.x = piv.x*inv; pinv.y = -piv.y*inv;
    for (int r = cc+1; r < 9; ++r){
      f2c fac = cmulf(M[r*9+cc], pinv);
      for (int t = cc+1; t < 9; ++t) M[r*9+t] = csubf(M[r*9+t], cmulf(fac, M[cc*9+t]));
    }
  }
  atomicAdd(&dlogJ[b], (double)logdet);   // global_atomic_add_f64
}

// ---------------- host ----------------
// d_in[0]: cfgs complex128 (interleaved doubles), d_in[1]: mask (unused — parity is
// recomputed analytically), d_in[2]: rho_root float64 (1,8,4).
// d_out (doubles): [ cfgs_out : CFG_DOUBLES ][ dlogJ : 4 ].
extern "C" void kernel_launch(void* const* d_in, const int* in_sizes, int n_in,
                              void* d_out, int out_size, void* d_ws, size_t ws_size,
                              hipStream_t stream)
{
  (void)in_sizes; (void)n_in; (void)out_size; (void)d_ws; (void)ws_size;
  const double* cfgs_in  = (const double*)d_in[0];
  const double* rho_root = (const double*)d_in[2];
  double* cfgs  = (double*)d_out;
  double* dlogJ = cfgs + CFG_DOUBLES;

  hipMemcpyAsync(cfgs, cfgs_in, (size_t)CFG_DOUBLES*sizeof(double),
                 hipMemcpyDeviceToDevice, stream);
  zero_dlogj<<<1, 32, 0, stream>>>(dlogJ);

  // 8 sequentially-dependent checkerboard sub-steps; state evolves in place in d_out.
  for (int mu = 0; mu < DD; ++mu)
    for (int eo = 0; eo < 2; ++eo)
      stout_step<<< (BSN*NSITE/2)/32, 32, 0, stream >>>(cfgs, dlogJ, rho_root, mu, eo);
}